// MultiBoxLoss_46127948759077
// MI455X (gfx1250) — compile-verified
//
#include <hip/hip_runtime.h>
#include <hip/hip_bf16.h>
#include <math.h>

// Problem constants (fixed by the reference)
constexpr int NB = 128;    // batch
constexpr int NA = 8732;   // anchors
constexpr int NC = 21;     // classes
constexpr int NM = 8;      // objects per image
constexpr int CE_BLOCKS = (NB * NA) / 256; // 4366, exact (no tail)
constexpr int TOPK_R = 35;                 // 35*256 = 8960 >= NA

typedef float v2f __attribute__((ext_vector_type(2)));
typedef float v8f __attribute__((ext_vector_type(8)));

// ---------------------------------------------------------------------------
// Full-wave32 sum reduction on the matrix pipe (V_WMMA_F32_16X16X4_F32).
// Stage 1: A = [x | 0], B = ones  ->  D[i][j] = x[i] + x[i+16]  (row sums,
//          replicated across all 16 columns).
// Stage 2: A = ones, B = pairs of D's VGPRs accumulated over 4 WMMAs ->
//          every element equals the sum over all 32 lanes. Exact f32 adds.
// Requires EXEC == all ones (call only from convergent code).
// ---------------------------------------------------------------------------
__device__ __forceinline__ float waveReduceWMMA(float x) {
  v2f a; a[0] = x; a[1] = 0.0f;
  v2f one; one[0] = 1.0f; one[1] = 1.0f;
  v8f c = {};
  c = __builtin_amdgcn_wmma_f32_16x16x4_f32(false, a, false, one, (short)0, c,
                                            false, false);
  v8f d = {};
  v2f bb;
  bb[0] = c[0]; bb[1] = c[1];
  d = __builtin_amdgcn_wmma_f32_16x16x4_f32(false, one, false, bb, (short)0, d,
                                            false, false);
  bb[0] = c[2]; bb[1] = c[3];
  d = __builtin_amdgcn_wmma_f32_16x16x4_f32(false, one, false, bb, (short)0, d,
                                            false, false);
  bb[0] = c[4]; bb[1] = c[5];
  d = __builtin_amdgcn_wmma_f32_16x16x4_f32(false, one, false, bb, (short)0, d,
                                            false, false);
  bb[0] = c[6]; bb[1] = c[7];
  d = __builtin_amdgcn_wmma_f32_16x16x4_f32(false, one, false, bb, (short)0, d,
                                            false, false);
  return d[0];
}

// ---------------------------------------------------------------------------
// Kernel 1: per-image matching. One block per batch row.
// ---------------------------------------------------------------------------
__global__ __launch_bounds__(256) void k_match(
    const float* __restrict__ boxes,      // [NB, NM, 4] xy pixel coords
    const int*   __restrict__ labels,     // [NB, NM]
    const float* __restrict__ anchors,    // [NA, 4] cxcy
    float* __restrict__ true_locs,        // [NB, NA, 4]
    int*   __restrict__ true_classes,     // [NB, NA]
    int*   __restrict__ n_pos)            // [NB]
{
  __shared__ float s_iou[NA];
  __shared__ unsigned char s_obj[NA];
  __shared__ unsigned long long s_key[NM];
  __shared__ float s_box[NM][4];
  __shared__ int s_lab[NM];
  __shared__ float s_red[8];

  const int b = blockIdx.x;
  const int tid = threadIdx.x;

  if (tid < NM * 4) ((float*)s_box)[tid] = boxes[b * NM * 4 + tid];
  if (tid < NM) { s_lab[tid] = labels[b * NM + tid]; s_key[tid] = 0ull; }
  __syncthreads();

  // Precompute scaled (/300) boxes and areas for IoU
  float sx1[NM], sy1[NM], sx2[NM], sy2[NM], sar[NM];
#pragma unroll
  for (int m = 0; m < NM; ++m) {
    sx1[m] = s_box[m][0] / 300.0f;
    sy1[m] = s_box[m][1] / 300.0f;
    sx2[m] = s_box[m][2] / 300.0f;
    sy2[m] = s_box[m][3] / 300.0f;
    sar[m] = (sx2[m] - sx1[m]) * (sy2[m] - sy1[m]);
  }

  // Per-object local best (argmax over anchors)
  float bi[NM]; unsigned int ba[NM];
#pragma unroll
  for (int m = 0; m < NM; ++m) { bi[m] = -1.0f; ba[m] = 0u; }

  for (int a = tid; a < NA; a += 256) {
    float cx = anchors[a * 4 + 0], cy = anchors[a * 4 + 1];
    float w  = anchors[a * 4 + 2], h  = anchors[a * 4 + 3];
    float ax1 = cx - w * 0.5f, ay1 = cy - h * 0.5f;
    float ax2 = cx + w * 0.5f, ay2 = cy + h * 0.5f;
    float areaB = (ax2 - ax1) * (ay2 - ay1);
    float best = -1.0f; int bm = 0;
#pragma unroll
    for (int m = 0; m < NM; ++m) {
      float wx = fminf(sx2[m], ax2) - fmaxf(sx1[m], ax1);
      float wy = fminf(sy2[m], ay2) - fmaxf(sy1[m], ay1);
      wx = fmaxf(wx, 0.0f); wy = fmaxf(wy, 0.0f);
      float inter = wx * wy;
      float iou = inter / (sar[m] + areaB - inter);
      if (iou > best) { best = iou; bm = m; }       // first-index tie-break
      if (iou > bi[m]) { bi[m] = iou; ba[m] = (unsigned int)a; }
    }
    s_iou[a] = best;
    s_obj[a] = (unsigned char)bm;
  }

  // Block-wide per-object argmax: key = (iou bits << 32) | ~anchor
  // (iou >= 0 so float bits are order-preserving; ~a prefers smaller index)
#pragma unroll
  for (int m = 0; m < NM; ++m) {
    if (bi[m] >= 0.0f) {
      unsigned long long key =
          (((unsigned long long)__float_as_uint(bi[m])) << 32) |
          (unsigned long long)(0xFFFFFFFFu - ba[m]);
      atomicMax(&s_key[m], key);
    }
  }
  __syncthreads();

  // Sequential override scatter (last object wins on duplicates)
  if (tid == 0) {
    for (int m = 0; m < NM; ++m) {
      unsigned int astar = 0xFFFFFFFFu - (unsigned int)(s_key[m] & 0xFFFFFFFFull);
      s_obj[astar] = (unsigned char)m;
      s_iou[astar] = 1.0f;
    }
  }
  __syncthreads();

  // Pass B: labels + encoded target locs (reference passes raw xy pixel
  // boxes straight into cxcy_to_gcxgcy — replicate literally)
  float cnt = 0.0f;
  for (int a = tid; a < NA; a += 256) {
    int m = (int)s_obj[a];
    float v = s_iou[a];
    int lab = (v < 0.5f) ? 0 : s_lab[m];
    true_classes[b * NA + a] = lab;
    if (lab != 0) cnt += 1.0f;
    float cx = anchors[a * 4 + 0], cy = anchors[a * 4 + 1];
    float w  = anchors[a * 4 + 2], h  = anchors[a * 4 + 3];
    float t0 = (s_box[m][0] - cx) / (w / 10.0f);
    float t1 = (s_box[m][1] - cy) / (h / 10.0f);
    float t2 = 5.0f * logf(s_box[m][2] / w);
    float t3 = 5.0f * logf(s_box[m][3] / h);
    float* tl = &true_locs[(b * NA + a) * 4];
    tl[0] = t0; tl[1] = t1; tl[2] = t2; tl[3] = t3;
  }

  float wsum = waveReduceWMMA(cnt);
  if ((tid & 31) == 0) s_red[tid >> 5] = wsum;
  __syncthreads();
  if (tid == 0) {
    float t = 0.0f;
    for (int i = 0; i < 8; ++i) t += s_red[i];
    n_pos[b] = (int)(t + 0.5f);
  }
}

// ---------------------------------------------------------------------------
// Kernel 2: CE per anchor + positive-CE / masked-L1 per-block partials.
// Grid = NB*NA/256 exactly (no tail -> WMMA reductions stay convergent).
// ---------------------------------------------------------------------------
__global__ __launch_bounds__(256) void k_ce(
    const float* __restrict__ locs,         // [NB, NA, 4]
    const float* __restrict__ scores,       // [NB, NA, NC]
    const float* __restrict__ true_locs,    // [NB, NA, 4]
    const int*   __restrict__ true_classes, // [NB, NA]
    float* __restrict__ ce_neg,             // [NB, NA]
    float* __restrict__ part_ce,            // [CE_BLOCKS]
    float* __restrict__ part_l1)            // [CE_BLOCKS]
{
  __shared__ float r0[8], r1[8];
  const int idx = blockIdx.x * 256 + threadIdx.x;  // always < NB*NA
  const float* s = &scores[(size_t)idx * NC];

  float mx = s[0];
#pragma unroll
  for (int c = 1; c < NC; ++c) mx = fmaxf(mx, s[c]);
  float se = 0.0f;
#pragma unroll
  for (int c = 0; c < NC; ++c) se += expf(s[c] - mx);
  float lse = mx + logf(se);

  int cls = true_classes[idx];
  float ce = lse - s[cls];
  bool pos = (cls != 0);
  ce_neg[idx] = pos ? 0.0f : ce;

  float cep = pos ? ce : 0.0f;
  float l1 = 0.0f;
  if (pos) {
    const float* pl = &locs[(size_t)idx * 4];
    const float* tl = &true_locs[(size_t)idx * 4];
#pragma unroll
    for (int k = 0; k < 4; ++k) l1 += fabsf(pl[k] - tl[k]);
  }

  float w0 = waveReduceWMMA(cep);
  float w1 = waveReduceWMMA(l1);
  int lane = threadIdx.x & 31, wid = threadIdx.x >> 5;
  if (lane == 0) { r0[wid] = w0; r1[wid] = w1; }
  __syncthreads();
  if (threadIdx.x == 0) {
    float a0 = 0.0f, a1 = 0.0f;
    for (int i = 0; i < 8; ++i) { a0 += r0[i]; a1 += r1[i]; }
    part_ce[blockIdx.x] = a0;
    part_l1[blockIdx.x] = a1;
  }
}

// ---------------------------------------------------------------------------
// Kernel 3: top-K sum per row via MSB-first radix selection on float bit
// patterns (CE >= 0 -> uint order == float order). Row lives in registers
// (256 threads x 35 values, zero-padded; padding is exact since CE >= 0).
// Finds t* = K-th largest, then sum = sum_{v > t*} v + (K - cnt_gt) * t*,
// which equals the reference's sorted top-K sum including ties.
// Count accumulation uses LDS float atomics on integer-valued floats
// (exact => order-independent => deterministic); value sum uses fixed-order
// per-wave slots.
// ---------------------------------------------------------------------------
__global__ __launch_bounds__(256) void k_topk(
    const float* __restrict__ ce_neg,  // [NB, NA]
    const int*   __restrict__ n_pos,   // [NB]
    float* __restrict__ hard)          // [NB]
{
  __shared__ float s_cnt[31];
  __shared__ float s_wc[8];
  __shared__ float s_ws[8];

  const int b = blockIdx.x;
  const int tid = threadIdx.x;

  unsigned int u[TOPK_R];
#pragma unroll
  for (int r = 0; r < TOPK_R; ++r) {
    int i = r * 256 + tid;
    u[r] = (i < NA) ? __float_as_uint(ce_neg[b * NA + i]) : 0u;
  }
  if (tid < 31) s_cnt[tid] = 0.0f;
  const int K = 3 * n_pos[b];
  const float Kf = (float)K;
  __syncthreads();

  // 31-step bitwise search for the K-th largest bit pattern (bit 31 = sign = 0)
  unsigned int prefix = 0u;
  for (int it = 0; it < 31; ++it) {
    unsigned int cand = prefix | (1u << (30 - it));
    float c = 0.0f;
#pragma unroll
    for (int r = 0; r < TOPK_R; ++r) c += (u[r] >= cand) ? 1.0f : 0.0f;
    float wsum = waveReduceWMMA(c);
    if ((tid & 31) == 0) atomicAdd(&s_cnt[it], wsum);
    __syncthreads();
    if (s_cnt[it] >= Kf) prefix = cand;  // uniform decision across the block
  }

  // Final pass: strict-greater count and sum
  float cg = 0.0f, sg = 0.0f;
#pragma unroll
  for (int r = 0; r < TOPK_R; ++r) {
    if (u[r] > prefix) { cg += 1.0f; sg += __uint_as_float(u[r]); }
  }
  float wc = waveReduceWMMA(cg);
  float ws = waveReduceWMMA(sg);
  if ((tid & 31) == 0) { s_wc[tid >> 5] = wc; s_ws[tid >> 5] = ws; }
  __syncthreads();
  if (tid == 0) {
    float tc = 0.0f, ts = 0.0f;
    for (int i = 0; i < 8; ++i) { tc += s_wc[i]; ts += s_ws[i]; }
    float tval = __uint_as_float(prefix);
    hard[b] = (K <= 0) ? 0.0f : (ts + (Kf - tc) * tval);
  }
}

// ---------------------------------------------------------------------------
// Kernel 4: deterministic fixed-order final reduction -> scalar loss.
// ---------------------------------------------------------------------------
__global__ __launch_bounds__(256) void k_final(
    const float* __restrict__ part_ce,
    const float* __restrict__ part_l1,
    const float* __restrict__ hard,
    const int*   __restrict__ n_pos,
    float* __restrict__ out)
{
  __shared__ float r[32];
  const int tid = threadIdx.x;
  float a0 = 0.0f, a1 = 0.0f, a2 = 0.0f, a3 = 0.0f;
  for (int i = tid; i < CE_BLOCKS; i += 256) { a0 += part_ce[i]; a1 += part_l1[i]; }
  for (int i = tid; i < NB; i += 256) { a2 += hard[i]; a3 += (float)n_pos[i]; }
  a0 = waveReduceWMMA(a0);
  a1 = waveReduceWMMA(a1);
  a2 = waveReduceWMMA(a2);
  a3 = waveReduceWMMA(a3);
  int lane = tid & 31, wid = tid >> 5;
  if (lane == 0) { r[wid] = a0; r[8 + wid] = a1; r[16 + wid] = a2; r[24 + wid] = a3; }
  __syncthreads();
  if (tid == 0) {
    float c0 = 0, c1 = 0, c2 = 0, c3 = 0;
    for (int i = 0; i < 8; ++i) {
      c0 += r[i]; c1 += r[8 + i]; c2 += r[16 + i]; c3 += r[24 + i];
    }
    float npt = c3;  // n_pos_total (>= NB, never zero)
    out[0] = (c2 + c0) / npt + c1 / (npt * 4.0f);
  }
}

// ---------------------------------------------------------------------------
// Launcher. Workspace layout (~25.7 MiB total):
//   true_locs float[NB*NA*4] | true_classes int[NB*NA] | ce_neg float[NB*NA]
//   | n_pos int[NB] | part_ce float[CE_BLOCKS] | part_l1 float[CE_BLOCKS]
//   | hard float[NB]
// ---------------------------------------------------------------------------
extern "C" void kernel_launch(void* const* d_in, const int* in_sizes, int n_in,
                              void* d_out, int out_size, void* d_ws, size_t ws_size,
                              hipStream_t stream) {
  (void)in_sizes; (void)n_in; (void)out_size; (void)ws_size;
  const float* predicted_locs   = (const float*)d_in[0];
  const float* predicted_scores = (const float*)d_in[1];
  const float* boxes            = (const float*)d_in[2];
  const int*   labels           = (const int*)d_in[3];
  const float* anchor_boxes     = (const float*)d_in[4];
  float* out = (float*)d_out;

  char* w = (char*)d_ws;
  size_t off = 0;
  float* true_locs = (float*)(w + off);  off += (size_t)NB * NA * 4 * sizeof(float);
  int* true_classes = (int*)(w + off);   off += (size_t)NB * NA * sizeof(int);
  float* ce_neg = (float*)(w + off);     off += (size_t)NB * NA * sizeof(float);
  int* n_pos = (int*)(w + off);          off += (size_t)NB * sizeof(int);
  float* part_ce = (float*)(w + off);    off += (size_t)CE_BLOCKS * sizeof(float);
  float* part_l1 = (float*)(w + off);    off += (size_t)CE_BLOCKS * sizeof(float);
  float* hard = (float*)(w + off);       off += (size_t)NB * sizeof(float);

  k_match<<<NB, 256, 0, stream>>>(boxes, labels, anchor_boxes,
                                  true_locs, true_classes, n_pos);
  k_ce<<<CE_BLOCKS, 256, 0, stream>>>(predicted_locs, predicted_scores,
                                      true_locs, true_classes, ce_neg,
                                      part_ce, part_l1);
  k_topk<<<NB, 256, 0, stream>>>(ce_neg, n_pos, hard);
  k_final<<<1, 256, 0, stream>>>(part_ce, part_l1, hard, n_pos, out);
}